// CorrLookup_28020366639462
// MI455X (gfx1250) — compile-verified
//
#include <hip/hip_runtime.h>

// RAFT-style correlation pyramid lookup for gfx1250 (MI455X).
// Memory-bound gather + lerp: no matrix structure (per-pixel weights), so
// WMMA/TDM do not apply. Optimized for the HBM/L2 path: non-temporal loads
// and stores (corr ~377MB touched once > 192MB L2), coalesced channel-major
// output stores, wave32-friendly 256-thread blocks.

#define BATCH 8
#define HH    96
#define WW    256
#define HW    (HH * WW)          // 24576
#define NPIX  (BATCH * HW)       // 196608
#define RAD   4
#define TAPS  9                  // 2*RAD + 1
#define NLVL  4
#define NCH   (NLVL * TAPS)      // 36

__global__ __launch_bounds__(256) void CorrLookup_28020366639462_kernel(
    const float* __restrict__ corr0,
    const float* __restrict__ corr1,
    const float* __restrict__ corr2,
    const float* __restrict__ corr3,
    const float* __restrict__ flow,
    float* __restrict__ out)
{
    const int n = blockIdx.x * blockDim.x + threadIdx.x;
    if (n >= NPIX) return;

    const int b  = n / HW;
    const int hw = n - b * HW;

    // flow layout (B, 2, H, W); disparity is channel 0.
    const float disp = flow[b * 2 * HW + hw];

    // output layout (B, 36, H, W)
    float* __restrict__ obase = out + (size_t)b * NCH * HW + hw;

    const float* __restrict__ corrs[NLVL] = {corr0, corr1, corr2, corr3};

#pragma unroll
    for (int lvl = 0; lvl < NLVL; ++lvl) {
        const int   Wi    = WW >> lvl;
        const float scale = 1.0f / (float)(1 << lvl);

        const float d    = disp * scale;          // >= 0
        const float x0f  = floorf(d);
        const float w    = d - x0f;               // shared frac weight for all taps
        const int   base = (int)x0f - RAD;        // index of v[0]; in [-4, Wi-5]

        const float* __restrict__ row = corrs[lvl] + (size_t)n * Wi;

        // Gather the 10-tap contiguous window with zeros padding.
        // Clamp the address (always in-bounds -> uniform unpredicated loads),
        // then select 0 for OOB taps — exactly the reference's clip + where.
        float v[TAPS + 1];
#pragma unroll
        for (int j = 0; j <= TAPS; ++j) {
            int idx  = base + j;
            int cidx = idx < 0 ? 0 : (idx > Wi - 1 ? Wi - 1 : idx);
            float val = __builtin_nontemporal_load(row + cidx);  // TH=NT: no reuse
            v[j] = ((unsigned)idx < (unsigned)Wi) ? val : 0.0f;
        }

        // out[k] = (1-w)*v[k] + w*v[k+1]  — lerp along the window.
        const float omw = 1.0f - w;
#pragma unroll
        for (int k = 0; k < TAPS; ++k) {
            float r = fmaf(omw, v[k], w * v[k + 1]);
            // channel c = lvl*TAPS + k; coalesced across lanes (contiguous in w)
            __builtin_nontemporal_store(r, obase + (size_t)(lvl * TAPS + k) * HW);
        }
    }
}

extern "C" void kernel_launch(void* const* d_in, const int* in_sizes, int n_in,
                              void* d_out, int out_size, void* d_ws, size_t ws_size,
                              hipStream_t stream)
{
    (void)in_sizes; (void)n_in; (void)out_size; (void)d_ws; (void)ws_size;

    const float* corr0 = (const float*)d_in[0];
    const float* corr1 = (const float*)d_in[1];
    const float* corr2 = (const float*)d_in[2];
    const float* corr3 = (const float*)d_in[3];
    const float* flow  = (const float*)d_in[4];
    float* out = (float*)d_out;

    const int threads = 256;                       // 8 wave32 waves per block
    const int blocks  = (NPIX + threads - 1) / threads;  // 768
    CorrLookup_28020366639462_kernel<<<blocks, threads, 0, stream>>>(
        corr0, corr1, corr2, corr3, flow, out);
}